// GATLayer_20401094656456
// MI455X (gfx1250) — compile-verified
//
#include <hip/hip_runtime.h>
#include <math.h>

typedef float v2f __attribute__((ext_vector_type(2)));
typedef float v8f __attribute__((ext_vector_type(8)));
typedef unsigned int u32x4 __attribute__((ext_vector_type(4)));
typedef int          i32x4 __attribute__((ext_vector_type(4)));
typedef int          i32x8 __attribute__((ext_vector_type(8)));

#define N_NODES 8192
#define F_IN    512
#define F_OUT   64
#define NEG_BIG (-9000000000000000.0f)
#define SLOPE   0.2f

#if defined(__has_builtin)
#  if __has_builtin(__builtin_amdgcn_tensor_load_to_lds)
#    define USE_TDM 1
#  else
#    define USE_TDM 0
#  endif
#else
#  define USE_TDM 0
#endif

__device__ __forceinline__ float lrelu(float x) { return x > 0.0f ? x : SLOPE * x; }

// ---------------------------------------------------------------------------
// Kernel 1: h = input @ W via v_wmma_f32_16x16x4_f32 (proven codegen).
// ---------------------------------------------------------------------------
__global__ __launch_bounds__(128) void k_gemm_h(const float* __restrict__ inp,
                                                const float* __restrict__ W,
                                                float* __restrict__ h) {
  __shared__ float At[16][66];
  __shared__ float Bt[64][66];
  const int r0   = blockIdx.x * 16;
  const int tid  = threadIdx.x;
  const int lane = tid & 31;
  const int wave = tid >> 5;

  v8f acc = {};
  const int arow = lane & 15;
  const int koff = (lane >> 4) * 2;
  const int bcol = wave * 16 + (lane & 15);

  for (int k0 = 0; k0 < F_IN; k0 += 64) {
    for (int i = 0; i < 8; ++i) {
      int e = tid + 128 * i;
      int r = e >> 6, k = e & 63;
      At[r][k] = inp[(size_t)(r0 + r) * F_IN + k0 + k];
    }
    for (int i = 0; i < 32; ++i) {
      int e = tid + 128 * i;
      int k = e >> 6, c = e & 63;
      Bt[c][k] = W[(size_t)(k0 + k) * F_OUT + c];
    }
    __syncthreads();
    for (int ks = 0; ks < 64; ks += 4) {
      v2f a, b;
      a[0] = At[arow][ks + koff];
      a[1] = At[arow][ks + koff + 1];
      b[0] = Bt[bcol][ks + koff];
      b[1] = Bt[bcol][ks + koff + 1];
      acc = __builtin_amdgcn_wmma_f32_16x16x4_f32(false, a, false, b,
                                                  (short)0, acc, false, false);
    }
    __syncthreads();
  }
  const int col   = wave * 16 + (lane & 15);
  const int rbase = (lane >> 4) * 8;
  for (int v = 0; v < 8; ++v)
    h[(size_t)(r0 + rbase + v) * F_OUT + col] = acc[v];
}

// ---------------------------------------------------------------------------
// Kernel 2: hT[c][j] = h[j][c] (one-time transpose, LDS-tiled, coalesced).
// ---------------------------------------------------------------------------
__global__ __launch_bounds__(256) void k_transpose(const float* __restrict__ h,
                                                   float* __restrict__ hT) {
  __shared__ __align__(16) float T[64][68];
  const int j0  = blockIdx.x * 64;
  const int tid = threadIdx.x;
#pragma unroll
  for (int i = 0; i < 4; ++i) {
    int q  = tid + 256 * i;
    int j  = q >> 4;
    int c0 = (q & 15) * 4;
    float4 hv = *(const float4*)(h + (size_t)(j0 + j) * F_OUT + c0);
    T[c0 + 0][j] = hv.x;
    T[c0 + 1][j] = hv.y;
    T[c0 + 2][j] = hv.z;
    T[c0 + 3][j] = hv.w;
  }
  __syncthreads();
  const int c  = tid >> 2;
  const int jb = (tid & 3) * 16;
  float4*      dst4 = (float4*)(hT + (size_t)c * N_NODES + j0 + jb);
  const float* srcT = &T[c][jb];
  dst4[0] = *(const float4*)(srcT + 0);
  dst4[1] = *(const float4*)(srcT + 4);
  dst4[2] = *(const float4*)(srcT + 8);
  dst4[3] = *(const float4*)(srcT + 12);
}

// ---------------------------------------------------------------------------
// Kernel 3: src = h@a1, dst = h@a2. One wave per row, shuffle reduction.
// ---------------------------------------------------------------------------
__global__ __launch_bounds__(256) void k_srcdst(const float* __restrict__ h,
                                                const float* __restrict__ a,
                                                float* __restrict__ src,
                                                float* __restrict__ dst) {
  const int row  = blockIdx.x * 8 + (threadIdx.x >> 5);
  const int lane = threadIdx.x & 31;
  float h0 = h[(size_t)row * F_OUT + lane];
  float h1 = h[(size_t)row * F_OUT + lane + 32];
  float s1 = h0 * a[lane]         + h1 * a[lane + 32];
  float s2 = h0 * a[F_OUT + lane] + h1 * a[F_OUT + lane + 32];
  for (int off = 16; off; off >>= 1) {
    s1 += __shfl_xor(s1, off, 32);
    s2 += __shfl_xor(s2, off, 32);
  }
  if (lane == 0) { src[row] = s1; dst[row] = s2; }
}

// ---------------------------------------------------------------------------
// Kernel 4 (fused flash-style GAT attention, TDM-fed):
// Block: 256 threads / 8 waves / 32 output rows; wave w: row-tile w>>2,
// col-tile w&3. Per 64-wide chunk the 64x64 h^T tile is DMA'd into LDS by the
// Tensor Data Mover (D# per CDNA5 ISA ch.8; pad feature reproduces the 68-float
// padded LDS stride), overlapping with logit/exp computation; tracked with
// TENSORcnt. adj is streamed once (256 MB HBM floor) with prefetch.
// ---------------------------------------------------------------------------
__global__ __launch_bounds__(256) void k_attn_fused(const int* __restrict__ adj,
                                                    const float* __restrict__ hT,
                                                    const float* __restrict__ src,
                                                    const float* __restrict__ dstv,
                                                    float* __restrict__ out) {
  __shared__ __align__(16) float P [32][68];  // exp tile [row][j]
  __shared__ __align__(16) float Ht[64][68];  // h^T tile [col][j] (stride 68)
  __shared__ __align__(16) float sF[32];      // flash rescale factors
  __shared__ float sSrc[32], sM[32], sS[32], sCmax[32], sPsum[32];

  const int r0   = blockIdx.x * 32;
  const int tid  = threadIdx.x;
  const int lane = tid & 31;
  const int wave = tid >> 5;
  const int rt   = wave >> 2;
  const int ct   = wave & 3;

  const int pr = tid >> 3;
  const int pc = (tid & 7) * 8;

  if (tid < 32) {
    sSrc[tid] = src[r0 + tid];
    sM[tid]   = -INFINITY;
    sS[tid]   = 0.0f;
  }
  __syncthreads();

  v8f acc = {};
  const int arow   = rt * 16 + (lane & 15);
  const int koff   = (lane >> 4) * 2;
  const int bcol   = ct * 16 + (lane & 15);
  const int rbase  = rt * 16 + (lane >> 4) * 8;
  const float my_src = sSrc[pr];
  const int* __restrict__ arow_ptr = adj + (size_t)(r0 + pr) * N_NODES + pc;

#if USE_TDM
  // Constant parts of the Tensor DMA Descriptor (D#), CDNA5 ISA §8.3/§8.4.
  const unsigned lds_ht = (unsigned)(uintptr_t)&Ht[0][0];   // LDS byte offset
  i32x8 g1;
  g1[0] = (2 << 16)      // data_size = 4 bytes
        | (1 << 20)      // pad_enable
        | (5 << 22)      // pad_interval: 2^(5+1) = 64 DWORDs between pads
        | (3 << 25);     // pad_amount:   3+1    = 4 DWORDs (68-float stride)
  g1[1] = (int)(8192u << 16);   // tensor_dim0[15:0] = 8192 (elements)
  g1[2] = (int)(64u   << 16);   // tensor_dim0 hi = 0 | tensor_dim1 = 64
  g1[3] = (int)(64u   << 16);   // tensor_dim1 hi = 0 | tile_dim0 = 64
  g1[4] = 64;                   // tile_dim1 = 64, tile_dim2 = 0
  g1[5] = 8192;                 // tensor_dim0_stride lo32 (elements)
  g1[6] = 0;
  g1[7] = 0;
  const i32x4 gz = {0, 0, 0, 0};
#endif

  for (int j0 = 0; j0 < N_NODES; j0 += 64) {
#if USE_TDM
    if (wave == 7) {            // one wave drives the DMA engine
      unsigned long long ga = (unsigned long long)(uintptr_t)hT
                            + (unsigned long long)j0 * 4ull;
      u32x4 g0;
      g0[0] = 1u;                                         // count=1 (valid)
      g0[1] = lds_ht;                                     // LDS dest (bytes)
      g0[2] = (unsigned)(ga & 0xffffffffu);               // global_addr lo
      g0[3] = (unsigned)((ga >> 32) & 0x1ffffffu)         // global_addr hi
            | (2u << 30);                                 // type = 2 (image)
#if __clang_major__ >= 23
      __builtin_amdgcn_tensor_load_to_lds(g0, g1, gz, gz, g1, 0);
#else
      __builtin_amdgcn_tensor_load_to_lds(g0, g1, gz, gz, 0);
#endif
    }
#endif

    // ---- Phase A: adj -> logits (regs) + chunk row-max ----
    int4 a0 = *(const int4*)(arow_ptr + j0);
    int4 a1 = *(const int4*)(arow_ptr + j0 + 4);
    if (j0 + 64 < N_NODES)
      __builtin_prefetch(arow_ptr + j0 + 64, 0, 0);
    float4 d0 = *(const float4*)(dstv + j0 + pc);
    float4 d1 = *(const float4*)(dstv + j0 + pc + 4);
    float ev[8];
    {
      int   av[8] = {a0.x, a0.y, a0.z, a0.w, a1.x, a1.y, a1.z, a1.w};
      float dd[8] = {d0.x, d0.y, d0.z, d0.w, d1.x, d1.y, d1.z, d1.w};
#pragma unroll
      for (int d = 0; d < 8; ++d)
        ev[d] = (av[d] > 0) ? lrelu(my_src + dd[d]) : NEG_BIG;
    }
    float cm = ev[0];
#pragma unroll
    for (int d = 1; d < 8; ++d) cm = fmaxf(cm, ev[d]);
    cm = fmaxf(cm, __shfl_xor(cm, 1, 32));
    cm = fmaxf(cm, __shfl_xor(cm, 2, 32));
    cm = fmaxf(cm, __shfl_xor(cm, 4, 32));
    if ((tid & 7) == 0) sCmax[pr] = cm;

#if USE_TDM
    if (wave == 7) __builtin_amdgcn_s_wait_tensorcnt(0);  // DMA done -> LDS
#else
#pragma unroll
    for (int i = 0; i < 4; ++i) {               // fallback: b128 copy
      int q  = tid + 256 * i;
      int c  = q >> 4;
      int jb = (q & 15) * 4;
      float4 hv = *(const float4*)(hT + (size_t)c * N_NODES + j0 + jb);
      *(float4*)&Ht[c][jb] = hv;
    }
#endif
    __syncthreads();

    // ---- Phase B: online max merge ----
    if (tid < 32) {
      float mo = sM[tid];
      float mn = fmaxf(mo, sCmax[tid]);
      sF[tid]  = __expf(mo - mn);
      sM[tid]  = mn;
    }
    __syncthreads();

    // ---- Phase C: P = exp(e - m_new), chunk sum, accumulator rescale ----
    {
      float mn = sM[pr];
      float pv[8];
      float ps = 0.0f;
#pragma unroll
      for (int d = 0; d < 8; ++d) { pv[d] = __expf(ev[d] - mn); ps += pv[d]; }
      *(float4*)&P[pr][pc]     = make_float4(pv[0], pv[1], pv[2], pv[3]);
      *(float4*)&P[pr][pc + 4] = make_float4(pv[4], pv[5], pv[6], pv[7]);
      ps += __shfl_xor(ps, 1, 32);
      ps += __shfl_xor(ps, 2, 32);
      ps += __shfl_xor(ps, 4, 32);
      if ((tid & 7) == 0) sPsum[pr] = ps;

      float4 f0 = *(const float4*)&sF[rbase];
      float4 f1 = *(const float4*)&sF[rbase + 4];
      acc[0] *= f0.x; acc[1] *= f0.y; acc[2] *= f0.z; acc[3] *= f0.w;
      acc[4] *= f1.x; acc[5] *= f1.y; acc[6] *= f1.z; acc[7] *= f1.w;
    }
    __syncthreads();

    // ---- Phase D: running-sum update (wave 0) overlapped with 16 WMMAs ----
    if (tid < 32) sS[tid] = sS[tid] * sF[tid] + sPsum[tid];
#pragma unroll
    for (int ks = 0; ks < 64; ks += 4) {
      v2f a, b;
      a[0] = P [arow][ks + koff];
      a[1] = P [arow][ks + koff + 1];
      b[0] = Ht[bcol][ks + koff];
      b[1] = Ht[bcol][ks + koff + 1];
      acc = __builtin_amdgcn_wmma_f32_16x16x4_f32(false, a, false, b,
                                                  (short)0, acc, false, false);
    }
    __syncthreads();
  }

  const int col = ct * 16 + (lane & 15);
#pragma unroll
  for (int v = 0; v < 8; ++v) {
    int   r = rbase + v;
    float x = acc[v] / sS[r];
    out[(size_t)(r0 + r) * F_OUT + col] = (x > 0.0f) ? x : (__expf(x) - 1.0f);
  }
}

// ---------------------------------------------------------------------------
extern "C" void kernel_launch(void* const* d_in, const int* in_sizes, int n_in,
                              void* d_out, int out_size, void* d_ws, size_t ws_size,
                              hipStream_t stream) {
  const float* inp = (const float*)d_in[0];
  const int*   adj = (const int*)  d_in[1];
  const float* W   = (const float*)d_in[2];
  const float* a   = (const float*)d_in[3];

  float* ws  = (float*)d_ws;
  float* h   = ws;                                   // 8192*64
  float* hT  = h   + (size_t)N_NODES * F_OUT;        // 64*8192
  float* src = hT  + (size_t)N_NODES * F_OUT;
  float* dst = src + N_NODES;
  float* out = (float*)d_out;

  k_gemm_h    <<<N_NODES / 16, 128, 0, stream>>>(inp, W, h);
  k_transpose <<<N_NODES / 64, 256, 0, stream>>>(h, hT);
  k_srcdst    <<<N_NODES / 8,  256, 0, stream>>>(h, a, src, dst);
  k_attn_fused<<<N_NODES / 32, 256, 0, stream>>>(adj, hT, src, dst, out);
}